// triplet_loss_52183852646786
// MI455X (gfx1250) — compile-verified
//
#include <hip/hip_runtime.h>
#include <math.h>

typedef __attribute__((ext_vector_type(2))) float v2f;
typedef __attribute__((ext_vector_type(8))) float v8f;

#define NROWS 8192
#define DIM   128
#define COL_SEGS 4                       // split column sweep for parallelism
#define SEG_COLS (NROWS / COL_SEGS)      // 2048 columns per segment
#define ALFA 1.0f

// ---------------- Phase 0: squared norms, one wave per row ----------------
__global__ void sqnorm_kernel(const float* __restrict__ emb, float* __restrict__ sq) {
    const int lane = threadIdx.x & 31;
    const int wave = threadIdx.x >> 5;
    const int row  = blockIdx.x * (blockDim.x >> 5) + wave;
    const float4* p = (const float4*)(emb + (size_t)row * DIM);
    float4 v = p[lane];                          // 32 lanes x float4 = 128 floats
    float s = v.x * v.x + v.y * v.y + v.z * v.z + v.w * v.w;
    #pragma unroll
    for (int m = 16; m >= 1; m >>= 1) s += __shfl_xor(s, m, 32);
    if (lane == 0) sq[row] = s;
}

// ---- Phase 1: fused gram(WMMA f32) -> dist -> batch-hard dp/dn per row ----
// One wave owns a 16-row tile; blockIdx.y selects a 2048-column segment.
__global__ void __launch_bounds__(256)
triplet_hard_kernel(const float* __restrict__ emb, const int* __restrict__ labels,
                    const float* __restrict__ sq,
                    float* __restrict__ dp_out, float* __restrict__ dn_out) {
    const int lane = threadIdx.x & 31;
    const int wave = threadIdx.x >> 5;
    const int tile = blockIdx.x * (blockDim.x >> 5) + wave;  // 16-row tile
    const int row_base = tile * 16;
    const int seg = blockIdx.y;
    const int half = lane >> 4;   // which 16-lane half
    const int sub  = lane & 15;

    // ---- preload A slab: 16 rows x 128 K, per-lane m=sub, K = k4*4 + 2*half + {0,1}
    v2f a[32];
    const float* arow = emb + (size_t)(row_base + sub) * DIM + 2 * half;
    #pragma unroll
    for (int k4 = 0; k4 < 32; ++k4)
        a[k4] = *(const v2f*)(arow + k4 * 4);

    // labels / sq-norms for the 8 rows this lane accumulates (m = r + 8*half)
    int   labr[8];
    float sqr[8];
    #pragma unroll
    for (int r = 0; r < 8; ++r) {
        const int m = r + 8 * half;
        labr[r] = labels[row_base + m];
        sqr[r]  = sq[row_base + m];
    }

    float dp[8], dn[8];
    #pragma unroll
    for (int r = 0; r < 8; ++r) { dp[r] = 0.0f; dn[r] = INFINITY; }

    const int col_begin = seg * SEG_COLS;
    const int col_end   = col_begin + SEG_COLS;

    for (int col_base = col_begin; col_base < col_end; col_base += 16) {
        // B tile 4x16 per k-step: lane = column n, same half/K striping as A
        const float* brow = emb + (size_t)(col_base + sub) * DIM + 2 * half;
        v8f c = {};
        #pragma unroll
        for (int k4 = 0; k4 < 32; ++k4) {
            v2f b = *(const v2f*)(brow + k4 * 4);
            // D = A(16x4,f32) * B(4x16,f32) + C  -> v_wmma_f32_16x16x4_f32
            c = __builtin_amdgcn_wmma_f32_16x16x4_f32(
                    false, a[k4], false, b, (short)0, c, false, false);
        }

        const int   j    = col_base + sub;       // column this lane holds
        const int   labc = labels[j];
        const float sqc  = sq[j];
        #pragma unroll
        for (int r = 0; r < 8; ++r) {
            const int   i    = row_base + r + 8 * half;
            const float d2   = sqr[r] + sqc - 2.0f * c[r];
            const float dist = sqrtf(fmaxf(d2, 0.0f) + 1e-12f);
            const bool  same = (labr[r] == labc);
            const bool  pos  = same && (i != j);
            dp[r] = fmaxf(dp[r], pos ? dist : 0.0f);      // furthest positive
            dn[r] = fminf(dn[r], same ? INFINITY : dist); // closest negative
        }
    }

    // reduce max/min across the 16 lanes of each half (masks stay in-half)
    #pragma unroll
    for (int r = 0; r < 8; ++r) {
        #pragma unroll
        for (int m = 8; m >= 1; m >>= 1) {
            dp[r] = fmaxf(dp[r], __shfl_xor(dp[r], m, 32));
            dn[r] = fminf(dn[r], __shfl_xor(dn[r], m, 32));
        }
    }
    if (sub == 0) {
        #pragma unroll
        for (int r = 0; r < 8; ++r) {
            const int i = row_base + r + 8 * half;
            dp_out[(size_t)seg * NROWS + i] = dp[r];
            dn_out[(size_t)seg * NROWS + i] = dn[r];
        }
    }
}

// ---------------- Phase 2: hinge + mean over all rows ----------------
__global__ void finalize_kernel(const float* __restrict__ dp_part,
                                const float* __restrict__ dn_part,
                                float* __restrict__ out) {
    __shared__ float red[256];
    float s = 0.0f;
    for (int i = threadIdx.x; i < NROWS; i += 256) {
        float dp = 0.0f, dn = INFINITY;
        #pragma unroll
        for (int g = 0; g < COL_SEGS; ++g) {
            dp = fmaxf(dp, dp_part[(size_t)g * NROWS + i]);
            dn = fminf(dn, dn_part[(size_t)g * NROWS + i]);
        }
        s += fmaxf(dp - dn + ALFA, 0.0f);     // dn==inf -> hinge 0, matches ref
    }
    red[threadIdx.x] = s;
    __syncthreads();
    for (int stride = 128; stride > 0; stride >>= 1) {
        if (threadIdx.x < stride) red[threadIdx.x] += red[threadIdx.x + stride];
        __syncthreads();
    }
    if (threadIdx.x == 0) out[0] = red[0] / (float)NROWS;
}

extern "C" void kernel_launch(void* const* d_in, const int* in_sizes, int n_in,
                              void* d_out, int out_size, void* d_ws, size_t ws_size,
                              hipStream_t stream) {
    const float* emb    = (const float*)d_in[0];   // [8192,128] f32
    const int*   labels = (const int*)d_in[1];     // [8192] int
    float* out = (float*)d_out;                    // scalar f32

    float* sq = (float*)d_ws;                                  // [8192]
    float* dp = sq + NROWS;                                    // [COL_SEGS][8192]
    float* dn = dp + (size_t)COL_SEGS * NROWS;                 // [COL_SEGS][8192]

    // Phase 0: squared norms (8 waves/block, one wave per row)
    sqnorm_kernel<<<NROWS / 8, 256, 0, stream>>>(emb, sq);

    // Phase 1: 512 row-tiles x 4 col-segments, 8 waves per block
    dim3 grid(NROWS / (16 * 8), COL_SEGS);
    triplet_hard_kernel<<<grid, 256, 0, stream>>>(emb, labels, sq, dp, dn);

    // Phase 2: mean hinge
    finalize_kernel<<<1, 256, 0, stream>>>(dp, dn, out);
}